// GAT_24017457119793
// MI455X (gfx1250) — compile-verified
//
#include <hip/hip_runtime.h>
#include <hip/hip_bf16.h>
#include <math.h>

// ---------------------------------------------------------------------------
// GAT (23 nodes, 1 feat, 2 hid, 2 heads) + MLP tail for MI455X (gfx1250).
// Kernel 1: per-sample GAT -> x92 workspace (row-major [B,92]).
// Kernel 2: fc1/pre via V_WMMA_F32_16X16X4_F32 (fp32 matrix path), then
//           predict/getvar per-lane. One wave32 per 16-sample tile.
// ---------------------------------------------------------------------------

typedef __attribute__((ext_vector_type(2))) float v2f;
typedef __attribute__((ext_vector_type(8))) float v8f;

#define BATCH    16384
#define NNODES   23
#define NADJ     (23 * 23)   // 529
#define XDIM     92          // 23 * 4 (2 hid * 2 heads)
#define FC1_OUT  128
#define PRE_OUT  16
#define NEG_BIG  -1e20f

__device__ __forceinline__ float gelu_exact(float v) {
    return 0.5f * v * (1.0f + erff(v * 0.70710678118654752f));
}

__device__ __forceinline__ float leaky02(float v) {
    return v > 0.0f ? v : 0.2f * v;
}

// ---------------------------------------------------------------------------
// Kernel 1: one thread per sample. Column-wise (axis=1) online softmax,
// both heads fused so adjacency is streamed only twice.
// ---------------------------------------------------------------------------
__global__ __launch_bounds__(256) void gat_kernel(
    const float* __restrict__ node_attr,   // [B,23,1]
    const int*   __restrict__ adj,         // [B,23,23]
    const float* __restrict__ W0, const float* __restrict__ a0,
    const float* __restrict__ W1, const float* __restrict__ a1,
    float* __restrict__ X)                 // [B,92]
{
    const int t = blockIdx.x * 256 + threadIdx.x;   // exact grid: 64*256=16384

    const float w00 = W0[0], w01 = W0[1];
    const float w10 = W1[0], w11 = W1[1];
    const float c1_0 = w00 * a0[0] + w01 * a0[1];
    const float c2_0 = w00 * a0[2] + w01 * a0[3];
    const float c1_1 = w10 * a1[0] + w11 * a1[1];
    const float c2_1 = w10 * a1[2] + w11 * a1[3];

    const float* xp = node_attr + t * NNODES;
    const int*   ap = adj + t * NADJ;

    float x[NNODES];
#pragma unroll
    for (int j = 0; j < NNODES; ++j) x[j] = xp[j];

    // Column softmax stats per head (register arrays via full unroll on j).
    float m0[NNODES], d0[NNODES], m1[NNODES], d1[NNODES];
#pragma unroll
    for (int j = 0; j < NNODES; ++j) {
        m0[j] = -1e30f; d0[j] = 0.0f;
        m1[j] = -1e30f; d1[j] = 0.0f;
    }

    // Pass 1: streaming max/denominator per column j (normalized over i).
    for (int i = 0; i < NNODES; ++i) {
        const float xi = xp[i];
        const int* arow = ap + i * NNODES;
        const float b0 = c1_0 * xi;
        const float b1 = c1_1 * xi;
#pragma unroll
        for (int j = 0; j < NNODES; ++j) {
            const int av = arow[j];
            float e0 = leaky02(b0 + c2_0 * x[j]);
            float e1 = leaky02(b1 + c2_1 * x[j]);
            e0 = (av > 0) ? e0 : NEG_BIG;
            e1 = (av > 0) ? e1 : NEG_BIG;
            float mn0 = fmaxf(m0[j], e0);
            d0[j] = d0[j] * __expf(m0[j] - mn0) + __expf(e0 - mn0);
            m0[j] = mn0;
            float mn1 = fmaxf(m1[j], e1);
            d1[j] = d1[j] * __expf(m1[j] - mn1) + __expf(e1 - mn1);
            m1[j] = mn1;
        }
    }
#pragma unroll
    for (int j = 0; j < NNODES; ++j) {
        d0[j] = 1.0f / d0[j];
        d1[j] = 1.0f / d1[j];
    }

    // Pass 2: s[i] = sum_j att[i][j] * x[j]; out = gelu(W[f]*s[i]).
    float* Xo = X + t * XDIM;
    for (int i = 0; i < NNODES; ++i) {
        const float xi = xp[i];
        const int* arow = ap + i * NNODES;
        const float b0 = c1_0 * xi;
        const float b1 = c1_1 * xi;
        float s0 = 0.0f, s1 = 0.0f;
#pragma unroll
        for (int j = 0; j < NNODES; ++j) {
            const int av = arow[j];
            float e0 = leaky02(b0 + c2_0 * x[j]);
            float e1 = leaky02(b1 + c2_1 * x[j]);
            e0 = (av > 0) ? e0 : NEG_BIG;
            e1 = (av > 0) ? e1 : NEG_BIG;
            s0 += __expf(e0 - m0[j]) * d0[j] * x[j];
            s1 += __expf(e1 - m1[j]) * d1[j] * x[j];
        }
        // concat order: [head0_f0, head0_f1, head1_f0, head1_f1]
        Xo[i * 4 + 0] = gelu_exact(w00 * s0);
        Xo[i * 4 + 1] = gelu_exact(w01 * s0);
        Xo[i * 4 + 2] = gelu_exact(w10 * s1);
        Xo[i * 4 + 3] = gelu_exact(w11 * s1);
    }
}

// ---------------------------------------------------------------------------
// Kernel 2: MLP tail. One wave32 handles a 16-row batch tile with fp32 WMMA.
//   fc1:  [16,92]x[92,128]  -> 8 N-tiles x 23 K-steps of wmma_f32_16x16x4_f32
//   pre:  [16,128]x[128,16] -> 1 N-tile  x 32 K-steps
// Fragment layout (ISA 7.12.2): A lane L holds A[L%16][4t+2*(L/16)+{0,1}];
// B lane L holds B[4t+2*(L/16)+{0,1}][L%16]; C/D lane L vgpr r holds
// C[r+8*(L/16)][L%16].
// ---------------------------------------------------------------------------
__global__ __launch_bounds__(256) void mlp_kernel(
    const float* __restrict__ X,           // [B,92] from kernel 1
    const float* __restrict__ fc1_w,       // [92,128]
    const float* __restrict__ fc1_b,       // [128]
    const float* __restrict__ pre_w,       // [128,16]
    const float* __restrict__ pre_b,       // [16]
    const float* __restrict__ predict_w,   // [16]
    const float* __restrict__ predict_b,   // [1]
    const float* __restrict__ getvar_w,    // [16]
    const float* __restrict__ getvar_b,    // [1]
    float* __restrict__ out)               // [y:16384][var:16384][xc:16384*128]
{
    __shared__ float sW1[XDIM * FC1_OUT];      // 47104 B
    __shared__ float sW2[FC1_OUT * PRE_OUT];   //  8192 B
    __shared__ float sXP[8][16 * 16];          //  8192 B

    // Cooperative weight staging (coalesced) into LDS.
    for (int idx = threadIdx.x; idx < XDIM * FC1_OUT; idx += 256)
        sW1[idx] = fc1_w[idx];
    for (int idx = threadIdx.x; idx < FC1_OUT * PRE_OUT; idx += 256)
        sW2[idx] = pre_w[idx];
    __syncthreads();

    const int lane = threadIdx.x & 31;
    const int wave = threadIdx.x >> 5;
    const int col16 = lane & 15;      // N/M index within 16
    const int hup   = lane >> 4;      // which K-half this lane covers
    const int row0  = (blockIdx.x * 8 + wave) * 16;   // 128 blocks * 8 waves * 16 = 16384

    float* out_y   = out;
    float* out_var = out + BATCH;
    float* out_xc  = out + 2 * BATCH;

    // ---------------- fc1: xc = relu(x92 @ W1 + b1) ----------------
    const v8f vzero = {0.f, 0.f, 0.f, 0.f, 0.f, 0.f, 0.f, 0.f};
    v8f acc[8];
#pragma unroll
    for (int nt = 0; nt < 8; ++nt) acc[nt] = vzero;

    const float* Arow = X + (row0 + col16) * XDIM;
    for (int t = 0; t < 23; ++t) {
        const int k = 4 * t + 2 * hup;
        v2f a;
        a.x = Arow[k];
        a.y = Arow[k + 1];
#pragma unroll
        for (int nt = 0; nt < 8; ++nt) {
            const int col = nt * 16 + col16;
            v2f b;
            b.x = sW1[k * FC1_OUT + col];
            b.y = sW1[(k + 1) * FC1_OUT + col];
            acc[nt] = __builtin_amdgcn_wmma_f32_16x16x4_f32(
                false, a, false, b, (short)0, acc[nt], false, false);
        }
    }

    // bias + relu + store xc (output #3), layout [B,128] row-major
#pragma unroll
    for (int nt = 0; nt < 8; ++nt) {
        const int col = nt * 16 + col16;
        const float bias = fc1_b[col];
#pragma unroll
        for (int r = 0; r < 8; ++r) {
            float v = acc[nt][r] + bias;
            v = v > 0.0f ? v : 0.0f;
            out_xc[(row0 + r + 8 * hup) * FC1_OUT + col] = v;
        }
    }
    // make xc visible before reloading it as the A operand of `pre`
    asm volatile("s_wait_storecnt 0" ::: "memory");

    // ---------------- pre: xp = gelu(xc @ W2 + b2) ----------------
    v8f pacc = vzero;
    const float* XCrow = out_xc + (row0 + col16) * FC1_OUT;
    for (int t = 0; t < 32; ++t) {
        const int k = 4 * t + 2 * hup;
        v2f a;
        a.x = XCrow[k];
        a.y = XCrow[k + 1];
        v2f b;
        b.x = sW2[k * PRE_OUT + col16];
        b.y = sW2[(k + 1) * PRE_OUT + col16];
        pacc = __builtin_amdgcn_wmma_f32_16x16x4_f32(
            false, a, false, b, (short)0, pacc, false, false);
    }

    const float pbias = pre_b[col16];
#pragma unroll
    for (int r = 0; r < 8; ++r) {
        float v = gelu_exact(pacc[r] + pbias);
        sXP[wave][(r + 8 * hup) * 16 + col16] = v;
    }
    __syncthreads();   // uniform across block; all WMMAs are done by now

    // ---------------- predict / getvar (16-wide dots) ----------------
    if (hup == 0) {
        float s = predict_b[0];
#pragma unroll
        for (int n = 0; n < 16; ++n)
            s += sXP[wave][col16 * 16 + n] * predict_w[n];
        out_y[row0 + col16] = s > 0.0f ? s : 0.0f;
    } else {
        float s = getvar_b[0];
#pragma unroll
        for (int n = 0; n < 16; ++n)
            s += sXP[wave][col16 * 16 + n] * getvar_w[n];
        out_var[row0 + col16] = s;
    }
}

// ---------------------------------------------------------------------------
extern "C" void kernel_launch(void* const* d_in, const int* in_sizes, int n_in,
                              void* d_out, int out_size, void* d_ws, size_t ws_size,
                              hipStream_t stream) {
    const float* node_attr = (const float*)d_in[0];
    const int*   adj       = (const int*)d_in[1];
    const float* W0        = (const float*)d_in[2];
    const float* a0        = (const float*)d_in[3];
    const float* W1        = (const float*)d_in[4];
    const float* a1        = (const float*)d_in[5];
    const float* fc1_w     = (const float*)d_in[6];
    const float* fc1_b     = (const float*)d_in[7];
    const float* pre_w     = (const float*)d_in[8];
    const float* pre_b     = (const float*)d_in[9];
    const float* predict_w = (const float*)d_in[10];
    const float* predict_b = (const float*)d_in[11];
    const float* getvar_w  = (const float*)d_in[12];
    const float* getvar_b  = (const float*)d_in[13];

    float* out = (float*)d_out;
    float* X   = (float*)d_ws;   // [16384, 92] fp32 scratch = 6.0 MB

    gat_kernel<<<BATCH / 256, 256, 0, stream>>>(node_attr, adj, W0, a0, W1, a1, X);
    mlp_kernel<<<BATCH / (8 * 16), 256, 0, stream>>>(X, fc1_w, fc1_b, pre_w, pre_b,
                                                     predict_w, predict_b,
                                                     getvar_w, getvar_b, out);
}